// MoDRouter_48507360641336
// MI455X (gfx1250) — compile-verified
//
#include <hip/hip_runtime.h>

// MoD router for MI455X (gfx1250, wave32).
// Stage 1: scores = hidden[16384 x 2048] @ gate[2048]  (bandwidth bound, 134 MB)
//   - coalesced float4 global loads -> LDS tiles
//   - per-lane partial sums arranged as WMMA 16x4 F32 A-matrix layout
//   - one v_wmma_f32_16x16x4_f32 with B=ones reduces 16 rows at once
// Stage 2: per-batch k-th largest threshold (k = S/2) via LDS bitonic sort,
//   mask = score >= threshold.

typedef __attribute__((ext_vector_type(2))) float v2f;
typedef __attribute__((ext_vector_type(8))) float v8f;

#define H_DIM 2048
#define CHUNK 128           // columns staged per iteration
#define ROW_PAD 132         // 128 + 4 floats pad to spread LDS banks
#define WAVES 8             // 256 threads / wave32
#define ROWS_PER_WAVE 16
#define ROWS_PER_BLOCK (WAVES * ROWS_PER_WAVE)   // 128

__global__ __launch_bounds__(256)
void mod_router_scores_kernel(const float* __restrict__ hidden,
                              const float* __restrict__ gate,
                              float* __restrict__ scores) {
    __shared__ float gate_s[H_DIM];
    __shared__ __align__(16) float tile[WAVES][ROWS_PER_WAVE][ROW_PAD];

    const int tid = threadIdx.x;
    // cache gate vector (8 KB) once per block
    for (int t = tid; t < H_DIM; t += 256) gate_s[t] = gate[t];
    __syncthreads();

    const int wave  = tid >> 5;
    const int lane  = tid & 31;
    const int m     = lane & 15;     // row within 16-row tile
    const int khalf = lane >> 4;     // 0: quarters 0,1  |  1: quarters 2,3
    const int q0    = khalf * 2;

    const long row0 = (long)blockIdx.x * ROWS_PER_BLOCK + (long)wave * ROWS_PER_WAVE;
    const float* hbase = hidden + row0 * H_DIM;

    float a0 = 0.0f, a1 = 0.0f;

    for (int c = 0; c < H_DIM / CHUNK; ++c) {
        const int base = c * CHUNK;
        // ---- coalesced load: 16 rows x 128 cols -> LDS (lane = float4 column) ----
        #pragma unroll
        for (int i = 0; i < ROWS_PER_WAVE; ++i) {
            const float4 v = *(const float4*)(hbase + (long)i * H_DIM + base + lane * 4);
            *(float4*)&tile[wave][i][lane * 4] = v;
        }
        __syncthreads();
        // ---- per-lane partial dot over this lane's two column-quarters ----
        #pragma unroll
        for (int j = 0; j < 8; ++j) {
            const float4 v = *(const float4*)&tile[wave][m][q0 * 32 + j * 4];
            const float4 g = *(const float4*)&gate_s[base + q0 * 32 + j * 4];
            a0 += v.x * g.x + v.y * g.y + v.z * g.z + v.w * g.w;
        }
        #pragma unroll
        for (int j = 0; j < 8; ++j) {
            const float4 v = *(const float4*)&tile[wave][m][(q0 + 1) * 32 + j * 4];
            const float4 g = *(const float4*)&gate_s[base + (q0 + 1) * 32 + j * 4];
            a1 += v.x * g.x + v.y * g.y + v.z * g.z + v.w * g.w;
        }
        __syncthreads();
    }

    // Lanes now hold the 16x4 F32 A-matrix of partial sums:
    //   lane L<16 : A[L][0]=a0, A[L][1]=a1 ; lane L>=16 : A[L-16][2]=a0, A[L-16][3]=a1
    // B = ones  =>  D[m][n] = sum_k A[m][k] = score[row0+m] for every n.
    v2f a;   a[0] = a0;   a[1] = a1;
    v2f one; one[0] = 1.0f; one[1] = 1.0f;
    v8f cacc = {};
    cacc = __builtin_amdgcn_wmma_f32_16x16x4_f32(
        /*neg_a=*/false, a, /*neg_b=*/false, one,
        /*c_mod=*/(short)0, cacc, /*reuse_a=*/false, /*reuse_b=*/false);

    // D layout: VGPR r = row r (lanes 0-15) / row r+8 (lanes 16-31), replicated over N.
    if (lane == 0) {
        #pragma unroll
        for (int r = 0; r < 8; ++r) scores[row0 + r] = cacc[r];
    } else if (lane == 16) {
        #pragma unroll
        for (int r = 0; r < 8; ++r) scores[row0 + 8 + r] = cacc[r];
    }
}

// One block per batch: bitonic sort 4096 scores in LDS, threshold = S-k-th
// ascending element, mask = score >= threshold (exactly k ones for unique thr).
__global__ __launch_bounds__(1024)
void mod_router_topk_mask_kernel(const float* __restrict__ scores,
                                 float* __restrict__ mask,
                                 int S, int k) {
    __shared__ float s[4096];
    const int b = blockIdx.x;
    const float* sc = scores + (long)b * S;

    for (int t = threadIdx.x; t < S; t += blockDim.x) s[t] = sc[t];
    __syncthreads();

    for (int kk = 2; kk <= S; kk <<= 1) {
        for (int j = kk >> 1; j > 0; j >>= 1) {
            for (int t = threadIdx.x; t < S; t += blockDim.x) {
                const int l = t ^ j;
                if (l > t) {
                    const bool up = ((t & kk) == 0);   // ascending block
                    const float x = s[t];
                    const float y = s[l];
                    if ((x > y) == up) { s[t] = y; s[l] = x; }
                }
            }
            __syncthreads();
        }
    }

    const float thr = s[S - k];   // k-th largest
    for (int t = threadIdx.x; t < S; t += blockDim.x)
        mask[(long)b * S + t] = (sc[t] >= thr) ? 1.0f : 0.0f;
}

extern "C" void kernel_launch(void* const* d_in, const int* in_sizes, int n_in,
                              void* d_out, int out_size, void* d_ws, size_t ws_size,
                              hipStream_t stream) {
    const float* hidden = (const float*)d_in[0];   // [B,S,H] f32
    const float* gate   = (const float*)d_in[1];   // [1,H]   f32
    float* out = (float*)d_out;                    // mask[B*S] ++ scores[B*S]

    const int H    = in_sizes[1];                  // 2048
    const int rows = in_sizes[0] / H;              // B*S = 16384
    const int B    = 4;
    const int S    = rows / B;                     // 4096
    const int k    = S / 2;                        // CAPACITY_GAMMA = 0.5

    float* mask_out   = out;
    float* scores_out = out + rows;

    (void)n_in; (void)out_size; (void)d_ws; (void)ws_size; (void)H;

    mod_router_scores_kernel<<<rows / ROWS_PER_BLOCK, 256, 0, stream>>>(
        hidden, gate, scores_out);
    mod_router_topk_mask_kernel<<<B, 1024, 0, stream>>>(
        scores_out, mask_out, S, k);
}